// SpanV2_73753178407290
// MI455X (gfx1250) — compile-verified
//
#include <hip/hip_runtime.h>
#include <stdint.h>

// ---------------------------------------------------------------------------
// Span-classifier head, fused:  logits = relu(gather(span)·W1 + b1)·W2 + b2
//  - f32 matrix pipe: V_WMMA_F32_16X16X4_F32 (exact f32 MAC)
//  - W1 chunks DMA'd global->LDS with GLOBAL_LOAD_ASYNC_TO_LDS_B128,
//    double-buffered, tracked with ASYNCcnt (overlaps DMA with WMMA)
//  - per-wave tile = 4 M-subtiles x 3 N-tiles: each B fragment feeds 4 WMMAs
//    from registers -> LDS issue rate ~0.4 ds/wmma (below LDS path limit)
// ---------------------------------------------------------------------------

typedef float v2f __attribute__((ext_vector_type(2)));
typedef float v8f __attribute__((ext_vector_type(8)));

#define HIDDEN        768
#define NUM_LABELS    57
#define WIDTH_DIM     150
#define IN_DIM        1686          // 768*2 + 150
#define SEQ           512
#define NSPANS_PER_B  5120
#define M_BLK         64            // spans per block
#define K_CHUNK       32            // K staging chunk (8 wmma k-steps)
#define NUM_CHUNKS    53            // ceil(1686/32) -> zero padded to 1696
#define BROW          772           // padded row stride (floats) for B/H in LDS
#define AROW          34            // padded (even!) row stride for A in LDS
#define BBUF_FLOATS   (K_CHUNK * BROW)   // one W1 staging buffer

// Issue one K-chunk of W1 as 12 async b128 DMAs per wave (512B each):
// 512 thr x 12 slots x 16B = 32x768x4B. Source row clamped for k>=IN_DIM
// (the A tile is zero there, so the product contributes exactly 0).
__device__ __forceinline__ void issue_w1_chunk_async(
    const float* __restrict__ W1, uint32_t lds_base, int cn, int t)
{
  const uint64_t w1base = (uint64_t)(uintptr_t)W1;
  #pragma unroll
  for (int i = 0; i < 12; ++i) {
    int fidx = t + i * 512;                 // 0..6143 float4 slots
    int row  = fidx / 192;                  // 192 float4 per 768-col row
    int col4 = fidx % 192;
    int k    = cn * K_CHUNK + row;
    int ksrc = (k < IN_DIM) ? k : 0;
    uint32_t goff = ((uint32_t)ksrc * HIDDEN + (uint32_t)col4 * 4u) * 4u;
    uint32_t loff = lds_base + ((uint32_t)row * BROW + (uint32_t)col4 * 4u) * 4u;
    asm volatile("global_load_async_to_lds_b128 %0, %1, %2"
                 :: "v"(loff), "v"(goff), "s"(w1base)
                 : "memory");
  }
}

__global__ __launch_bounds__(512) void span_head_fused(
    const float* __restrict__ hs,        // (8,512,768)
    const long long* __restrict__ spans, // (8,5120,3) int64
    const float* __restrict__ wemb,      // (11,150)
    const float* __restrict__ W1,        // (1686,768)
    const float* __restrict__ b1,        // (768)
    const float* __restrict__ W2,        // (768,57)
    const float* __restrict__ b2v,       // (57)
    float* __restrict__ out)             // (8,5120,57)
{
  extern __shared__ float smem[];
  // Scalar LDS-derived pointers only (aggregates holding the LDS symbol
  // become addrspacecast static initializers that ld.lld rejects).
  float* Bsm0 = smem;                        // 32 x 772
  float* Bsm1 = smem + BBUF_FLOATS;          // 32 x 772
  float* Asm  = smem + 2 * BBUF_FLOATS;      // 64 x 34
  float* Hsm  = smem;                        // 64 x 772 (aliases staging)

  // LDS byte offset of the dynamic-shared base: flat->LDS mapping keeps
  // addr[31:0] as the LDS byte address (ISA 10.2 aperture rules).
  const uint32_t smem_base = (uint32_t)(uintptr_t)smem;
  const uint32_t bbase0 = smem_base;
  const uint32_t bbase1 = smem_base + (uint32_t)(BBUF_FLOATS * 4);

  const int t    = threadIdx.x;
  const int w    = t >> 5;                // wave id 0..15
  const int lane = t & 31;
  const int lo   = lane & 15;             // tile row/col index
  const int hi   = lane >> 4;             // half-wave

  const int g0 = blockIdx.x * M_BLK;      // first span row of this block
  const int b  = g0 / NSPANS_PER_B;       // batch (5120 % 64 == 0 -> uniform)

  // GEMM1: every wave covers all 4 M-subtiles; N-tiles w, w+16, w+32.
  v8f acc[12] = {};                       // acc[m*3 + j]

  // Prime the pipeline: chunk 0 -> buffer 0
  issue_w1_chunk_async(W1, bbase0, 0, t);

  for (int c = 0; c < NUM_CHUNKS; ++c) {
    __syncthreads();   // compute c-1 done: buf[(c+1)&1] and Asm are free

    const bool has_next = (c + 1) < NUM_CHUNKS;
    if (has_next)
      issue_w1_chunk_async(W1, ((c + 1) & 1) ? bbase1 : bbase0, c + 1, t);

    // ---- gather span_emb chunk (M_BLK x K_CHUNK) into LDS (sync ds) ----
    #pragma unroll
    for (int i = 0; i < 4; ++i) {
      int idx = t + i * 512;              // 0..2047
      int m   = idx >> 5;                 // span row within block
      int kk  = idx & 31;
      int k   = c * K_CHUNK + kk;
      const long long* sp = spans + (long long)(g0 + m) * 3;
      float val = 0.f;
      if (k < HIDDEN) {
        int s = (int)sp[0];
        val = hs[((size_t)(b * SEQ + s)) * HIDDEN + k];
      } else if (k < 2 * HIDDEN) {
        int e = (int)sp[1];
        val = hs[((size_t)(b * SEQ + e)) * HIDDEN + (k - HIDDEN)];
      } else {
        int wk = k - 2 * HIDDEN;
        if (wk < WIDTH_DIM) {
          int wd = (int)sp[2];
          val = wemb[wd * WIDTH_DIM + wk];
        }
      }
      Asm[m * AROW + kk] = val;
    }

    // Retire chunk c's 12 DMAs (oldest; async loads complete in order).
    if (has_next)
      asm volatile("s_wait_asynccnt 0xc" ::: "memory");
    else
      asm volatile("s_wait_asynccnt 0x0" ::: "memory");
    __syncthreads();   // chunk c B data + A tile visible to all waves

    // ---- 8 k-steps of f32 WMMA over the chunk ----
    const float* Bsm  = (c & 1) ? Bsm1 : Bsm0;
    const float* acol = Asm + lo * AROW;          // row = lane%16
    #pragma unroll
    for (int ks = 0; ks < K_CHUNK; ks += 4) {
      const int kk0 = ks + 2 * hi;                // K = vgpr + 2*(lane/16)
      // 4 A frags (one per M-subtile): aligned ds_load_b64 each
      v2f a0 = *(const v2f*)(acol + 0 * 16 * AROW + kk0);
      v2f a1 = *(const v2f*)(acol + 1 * 16 * AROW + kk0);
      v2f a2 = *(const v2f*)(acol + 2 * 16 * AROW + kk0);
      v2f a3 = *(const v2f*)(acol + 3 * 16 * AROW + kk0);
      #pragma unroll
      for (int j = 0; j < 3; ++j) {
        int ncol = (w + 16 * j) * 16 + lo;
        v2f bf;   // B frag: col = lane%16, K = vgpr + 2*(lane/16)
        bf.x = Bsm[kk0 * BROW + ncol];
        bf.y = Bsm[(kk0 + 1) * BROW + ncol];
        // each B fragment feeds 4 WMMAs from registers
        acc[0 * 3 + j] = __builtin_amdgcn_wmma_f32_16x16x4_f32(
            false, a0, false, bf, (short)0, acc[0 * 3 + j], false, false);
        acc[1 * 3 + j] = __builtin_amdgcn_wmma_f32_16x16x4_f32(
            false, a1, false, bf, (short)0, acc[1 * 3 + j], false, false);
        acc[2 * 3 + j] = __builtin_amdgcn_wmma_f32_16x16x4_f32(
            false, a2, false, bf, (short)0, acc[2 * 3 + j], false, false);
        acc[3 * 3 + j] = __builtin_amdgcn_wmma_f32_16x16x4_f32(
            false, a3, false, bf, (short)0, acc[3 * 3 + j], false, false);
      }
    }
  }

  __syncthreads();   // staging LDS dead; reuse as H buffer

  // ---- bias + ReLU, write h (64 x 768) to LDS ----
  #pragma unroll
  for (int j = 0; j < 3; ++j) {
    int n = (w + 16 * j) * 16 + lo;
    float bias = b1[n];
    #pragma unroll
    for (int ms = 0; ms < 4; ++ms) {
      #pragma unroll
      for (int v = 0; v < 8; ++v) {
        int m = ms * 16 + v + 8 * hi;     // C layout: M = vgpr + 8*(lane/16)
        float hv = acc[ms * 3 + j][v] + bias;
        Hsm[m * BROW + n] = hv > 0.f ? hv : 0.f;
      }
    }
  }
  __syncthreads();

  // ---- GEMM2: (64 x 768) x (768 x 57), N padded to 64; one tile per wave ----
  const int msub2 = w >> 2;
  const int nt2   = w & 3;
  const int n2    = nt2 * 16 + lo;
  const bool nvalid = n2 < NUM_LABELS;
  v8f acc2 = {};
  const float* hrow = Hsm + (msub2 * 16 + lo) * BROW;   // BROW even -> b64 ok
  #pragma unroll 4
  for (int k = 0; k < HIDDEN; k += 4) {
    v2f a = *(const v2f*)(hrow + k + 2 * hi);
    int kk0 = k + 2 * hi;
    v2f bf;
    bf.x = nvalid ? W2[kk0 * NUM_LABELS + n2] : 0.f;
    bf.y = nvalid ? W2[(kk0 + 1) * NUM_LABELS + n2] : 0.f;
    acc2 = __builtin_amdgcn_wmma_f32_16x16x4_f32(
        false, a, false, bf, (short)0, acc2, false, false);
  }

  if (nvalid) {
    float bias = b2v[n2];
    #pragma unroll
    for (int v = 0; v < 8; ++v) {
      int m = msub2 * 16 + v + 8 * hi;
      out[(size_t)(g0 + m) * NUM_LABELS + n2] = acc2[v] + bias;
    }
  }
}

extern "C" void kernel_launch(void* const* d_in, const int* in_sizes, int n_in,
                              void* d_out, int out_size, void* d_ws, size_t ws_size,
                              hipStream_t stream) {
  const float*     hs    = (const float*)d_in[0];
  const long long* spans = (const long long*)d_in[1];
  const float*     wemb  = (const float*)d_in[2];
  const float*     W1    = (const float*)d_in[3];
  const float*     b1    = (const float*)d_in[4];
  const float*     W2    = (const float*)d_in[5];
  const float*     b2    = (const float*)d_in[6];
  float* out = (float*)d_out;

  const int n_spans = in_sizes[1] / 3;               // 40960
  const int grid    = n_spans / M_BLK;               // 640 blocks

  // LDS: max(double-buffered W1 staging + A tile, H buffer)
  const size_t stage = (size_t)(2 * BBUF_FLOATS + M_BLK * AROW) * sizeof(float);
  const size_t hbuf  = (size_t)M_BLK * BROW * sizeof(float);
  const size_t shmem = stage > hbuf ? stage : hbuf;  // ~206 KB < 320 KB WGP LDS

  (void)hipFuncSetAttribute((const void*)span_head_fused,
                            hipFuncAttributeMaxDynamicSharedMemorySize, (int)shmem);

  span_head_fused<<<grid, 512, shmem, stream>>>(hs, spans, wemb, W1, b1, W2, b2, out);
}